// VectorQuantizer_25649544691966
// MI455X (gfx1250) — compile-verified
//
#include <hip/hip_runtime.h>
#include <hip/hip_bf16.h>
#include <math.h>

typedef __attribute__((ext_vector_type(2))) float v2f;
typedef __attribute__((ext_vector_type(8))) float v8f;

#define N_TOK   131072      // 32 * 64 * 64 tokens
#define HW      4096        // 64 * 64
#define DDIM    64
#define KCODES  512
#define WGWAVES 8
#define NWG     1024        // N_TOK / (WGWAVES*16)
#define QELEMS  8388608     // N_TOK * DDIM

// ---------------- init: zero the histogram ----------------
__global__ void vq_init_kernel(unsigned int* __restrict__ cnt) {
    int t = threadIdx.x;
    if (t < KCODES) cnt[t] = 0u;
}

// ---------------- main: distances (WMMA f32) + argmin + gather + loss partials ----------------
__global__ __launch_bounds__(256) void vq_main_kernel(
        const float* __restrict__ in,       // [32,64,64,64] NCHW flat
        const float* __restrict__ w,        // [512,64]
        float* __restrict__ out,            // [N_TOK,64] quantized (flat)
        unsigned int* __restrict__ cnt,     // [512] histogram
        float* __restrict__ partials)       // [NWG*WGWAVES] loss partial sums
{
    // Codebook in B-operand layout: [code_tile][k_chunk][lane] -> float2
    __shared__ v2f   sB[32][16][32];        // 128 KB
    __shared__ float sWsq[KCODES];          // 2 KB
    __shared__ int   sIdx[WGWAVES][16];     // 512 B

    const int tid  = threadIdx.x;
    const int lane = tid & 31;
    const int wave = tid >> 5;

    // --- stage ||w_k||^2 (2 codes per thread) ---
    for (int i = 0; i < 2; ++i) {
        int code = tid * 2 + i;
        const float* wr = w + code * DDIM;
        float s = 0.f;
        #pragma unroll
        for (int j = 0; j < DDIM; ++j) s = fmaf(wr[j], wr[j], s);
        sWsq[code] = s;
    }

    // --- stage codebook into LDS, pre-swizzled to the WMMA B layout ---
    // B[k][n] = w[ct*16+n][k];  lane<16: (k=4kc+0,4kc+1, n=lane)  lane>=16: (k=4kc+2,4kc+3, n=lane-16)
    for (int e = tid; e < 32 * 16 * 32; e += 256) {
        int ct  = e >> 9;
        int rem = e & 511;
        int kc  = rem >> 5;
        int l   = rem & 31;
        int n   = l & 15;
        int kk  = kc * 4 + ((l >> 4) << 1);
        const float* wr = w + (ct * 16 + n) * DDIM + kk;
        v2f v; v.x = wr[0]; v.y = wr[1];
        sB[ct][kc][l] = v;
    }
    __syncthreads();

    // --- load A tile: 16 tokens x 64 dims in WMMA A layout (16 chunks of 16x4) ---
    const int tokBase = (blockIdx.x * WGWAVES + wave) * 16;
    const int m   = lane & 15;                    // token row within tile
    const int tok = tokBase + m;
    const int b   = tok >> 12;                    // tok / HW
    const int hw  = tok & (HW - 1);
    const float* inTok = in + (size_t)b * (DDIM * HW) + hw;
    const int cb = (lane >> 4) << 1;              // 0 for lanes 0-15, 2 for 16-31

    v2f A[16];
    #pragma unroll
    for (int kc = 0; kc < 16; ++kc) {
        A[kc].x = inTok[(size_t)(4 * kc + cb) * HW];
        A[kc].y = inTok[(size_t)(4 * kc + cb + 1) * HW];
    }

    // --- distance proxy (||w||^2 - 2 x.w) via chained f32 WMMA, running argmin ---
    float bestV[8];
    int   bestI[8];
    #pragma unroll
    for (int r = 0; r < 8; ++r) { bestV[r] = INFINITY; bestI[r] = 0; }

    for (int ct = 0; ct < 32; ++ct) {
        float wsq = sWsq[ct * 16 + m];
        v8f acc = {0.f, 0.f, 0.f, 0.f, 0.f, 0.f, 0.f, 0.f};
        #pragma unroll
        for (int kc = 0; kc < 16; ++kc) {
            v2f bv = sB[ct][kc][lane];
            acc = __builtin_amdgcn_wmma_f32_16x16x4_f32(
                    false, A[kc], false, bv, (short)0, acc, false, false);
        }
        int idx = ct * 16 + m;
        #pragma unroll
        for (int r = 0; r < 8; ++r) {
            float dv = fmaf(-2.f, acc[r], wsq);
            if (dv < bestV[r]) { bestV[r] = dv; bestI[r] = idx; }
        }
    }

    // --- argmin across the 16 columns held by the 16-lane halves (first-index tiebreak) ---
    #pragma unroll
    for (int msk = 8; msk >= 1; msk >>= 1) {
        #pragma unroll
        for (int r = 0; r < 8; ++r) {
            float ov = __shfl_xor(bestV[r], msk, 32);
            int   oi = __shfl_xor(bestI[r], msk, 32);
            if (ov < bestV[r] || (ov == bestV[r] && oi < bestI[r])) {
                bestV[r] = ov; bestI[r] = oi;
            }
        }
    }
    // lane 0 holds rows 0..7, lane 16 holds rows 8..15
    if (m == 0) {
        int rowOff = (lane >> 4) << 3;
        #pragma unroll
        for (int r = 0; r < 8; ++r) {
            sIdx[wave][rowOff + r] = bestI[r];
            atomicAdd(&cnt[bestI[r]], 1u);
        }
    }
    __syncthreads();

    // --- gather codebook rows -> output, fused loss accumulation (same flat offsets) ---
    float lsum = 0.f;
    for (int tk = 0; tk < 16; ++tk) {
        int code  = sIdx[wave][tk];
        int tokg  = tokBase + tk;
        v2f qv = ((const v2f*)(w + code * DDIM))[lane];      // 32 lanes x 2 = one 64-f row
        v2f iv = ((const v2f*)in)[(size_t)tokg * 32 + lane]; // contiguous flat input
        ((v2f*)out)[(size_t)tokg * 32 + lane] = qv;
        float dx = qv.x - iv.x;
        float dy = qv.y - iv.y;
        lsum = fmaf(dx, dx, fmaf(dy, dy, lsum));
    }
    #pragma unroll
    for (int msk = 16; msk >= 1; msk >>= 1)
        lsum += __shfl_xor(lsum, msk, 32);
    if (lane == 0)
        partials[blockIdx.x * WGWAVES + wave] = lsum;
}

// ---------------- finalize: deterministic loss reduction + perplexity ----------------
__global__ void vq_final_kernel(const unsigned int* __restrict__ cnt,
                                const float* __restrict__ partials,
                                float* __restrict__ out)
{
    __shared__ float red[512];
    int t = threadIdx.x;

    float s = 0.f;
    for (int i = t; i < NWG * WGWAVES; i += 512) s += partials[i];
    red[t] = s;
    __syncthreads();
    for (int st = 256; st >= 1; st >>= 1) {
        if (t < st) red[t] += red[t + st];
        __syncthreads();
    }
    float lossSum = red[0];
    __syncthreads();

    float p = (float)cnt[t] / (float)N_TOK;
    red[t] = p * logf(p + 1e-10f);
    __syncthreads();
    for (int st = 256; st >= 1; st >>= 1) {
        if (t < st) red[t] += red[t + st];
        __syncthreads();
    }
    if (t == 0) {
        out[QELEMS]     = 1.25f * lossSum / (float)QELEMS;  // q_loss + 0.25*e_loss
        out[QELEMS + 1] = expf(-red[0]);                    // perplexity
    }
}

extern "C" void kernel_launch(void* const* d_in, const int* in_sizes, int n_in,
                              void* d_out, int out_size, void* d_ws, size_t ws_size,
                              hipStream_t stream) {
    const float* in = (const float*)d_in[0];   // (32,64,64,64) f32
    const float* w  = (const float*)d_in[1];   // (512,64)      f32
    float* out = (float*)d_out;                // 8388608 + 2 f32

    unsigned int* cnt = (unsigned int*)d_ws;                     // 512 * u32
    float* partials   = (float*)((char*)d_ws + 2048);            // 8192 * f32

    vq_init_kernel<<<1, 512, 0, stream>>>(cnt);
    vq_main_kernel<<<NWG, 256, 0, stream>>>(in, w, out, cnt, partials);
    vq_final_kernel<<<1, 512, 0, stream>>>(cnt, partials, out);
}